// CrossAnalysisModel_60318520705759
// MI455X (gfx1250) — compile-verified
//
#include <hip/hip_runtime.h>

#define HID 64
#define NUSR 100000
#define NPOI 50000
#define NCAT 1000
#define E_UP 800000
#define E_PC 200000

typedef float v2f __attribute__((ext_vector_type(2)));
typedef float v8f __attribute__((ext_vector_type(8)));

// ---------------------------------------------------------------- utilities

__global__ void fill_u32_kernel(unsigned* __restrict__ p, unsigned bits, int n) {
  int i = blockIdx.x * blockDim.x + threadIdx.x;
  if (i < n) p[i] = bits;
}

__device__ __forceinline__ void atomicMaxFloat(float* addr, float value) {
  // monotonic bit-pattern trick; memory initialized to 0xFF800000 (-inf)
  if (value >= 0.0f)
    atomicMax((int*)addr, __float_as_int(value));
  else
    atomicMin((unsigned int*)addr, __float_as_uint(value));
}

// --------------------------------------------------- layer-0 node embedding
// out[n,f] = relu(x[n,:] @ W + b)[f] + emb[n,f]   (fin in {1,2,3})
__global__ void node_embed_kernel(const float* __restrict__ x, const float* __restrict__ W,
                                  const float* __restrict__ b, const float* __restrict__ emb,
                                  float* __restrict__ out, int n, int fin) {
  int i = blockIdx.x * blockDim.x + threadIdx.x;
  if (i >= n * HID) return;
  int node = i >> 6;
  int f = i & 63;
  float s = b[f];
  for (int k = 0; k < fin; ++k) s += x[node * fin + k] * W[k * HID + f];
  s = fmaxf(s, 0.0f);
  out[i] = s + emb[i];
}

// ------------------------------------------------------- fp32 WMMA GEMM
// out[N,HC] = X[N,K] @ W[K,HC] + bias[HC]
// Block = 256 threads = 8 waves; block owns a 128-row slab; wave w owns rows
// [block*128 + w*16, +16) and ALL HC columns (NT = HC/16 accumulator tiles).
// W (K*HC*4 = 32KB) + bias staged into LDS once per block via async LDS loads.
// Per k-step: one b64 A load per lane + NT LDS b32-pair B loads + NT WMMAs.
// Layouts per CDNA5 ISA 7.12.2:
//   A 16x4:  vgpr v, lanes[l16,hi] -> A[M=l16][K = v + 2*hi]
//   B 4x16:  vgpr v, lanes[l16,hi] -> B[K = v + 2*hi][N = l16]
//   C 16x16: vgpr v, lanes[l16,hi] -> C[M = v + 8*hi][N = l16]
template <int K, int HC>
__global__ __launch_bounds__(256) void gemm_bias_wmma_kernel(
    const float* __restrict__ X, const float* __restrict__ W,
    const float* __restrict__ bias, float* __restrict__ out, int N) {
  constexpr int NT = HC / 16;
  __shared__ __align__(16) float sW[K * HC];
  __shared__ __align__(16) float sB[HC];

  // ---- stage W and bias into LDS (gfx1250 async global->LDS, ASYNCcnt)
  for (int j = threadIdx.x; j < (K * HC) / 4; j += 256) {
    unsigned lds = (unsigned)(size_t)(sW + 4 * j);
    const float* g = W + 4 * j;
    asm volatile("global_load_async_to_lds_b128 %0, %1, off" ::"v"(lds), "v"(g) : "memory");
  }
  if (threadIdx.x < HC) {
    unsigned lds = (unsigned)(size_t)(sB + threadIdx.x);
    const float* g = bias + threadIdx.x;
    asm volatile("global_load_async_to_lds_b32 %0, %1, off" ::"v"(lds), "v"(g) : "memory");
  }
  asm volatile("s_wait_asynccnt 0x0" ::: "memory");
  __syncthreads();

  const int lane = threadIdx.x & 31;
  const int l16  = lane & 15;
  const int hi   = lane >> 4;
  const int m0   = (blockIdx.x * 8 + (threadIdx.x >> 5)) * 16;

  // Branchless edge handling: clamp the A row. A row M only contributes to
  // output row M, and out-of-range rows are never stored, so garbage is safe.
  const int arow = min(m0 + l16, N - 1);
  const float* xrow = X + (size_t)arow * K;

  v8f acc[NT];
#pragma unroll
  for (int t = 0; t < NT; ++t) acc[t] = (v8f){};

  for (int k = 0; k < K; k += 4) {
    const int kk = k + 2 * hi;
    const v2f a = *(const v2f*)(xrow + kk);  // 8B-aligned: kk even, K multiple of 4
#pragma unroll
    for (int t = 0; t < NT; ++t) {
      v2f b;
      b.x = sW[kk * HC + t * 16 + l16];
      b.y = sW[(kk + 1) * HC + t * 16 + l16];
      acc[t] = __builtin_amdgcn_wmma_f32_16x16x4_f32(false, a, false, b, (short)0, acc[t],
                                                     false, false);
    }
  }

  if (m0 >= N) return;  // whole-wave skip (after all WMMAs; EXEC was all-1s)
  const bool full = (m0 + 15) < N;
#pragma unroll
  for (int t = 0; t < NT; ++t) {
    const float bv = sB[t * 16 + l16];
#pragma unroll
    for (int v = 0; v < 8; ++v) {
      const int r = m0 + v + 8 * hi;
      if (full || r < N) out[(size_t)r * HC + t * 16 + l16] = acc[t][v] + bv;
    }
  }
}

// ----------------------------------------------------------- edge kernels
// One wave per (edge, head). C = 64 columns = 32 lanes x float2.

__global__ void edge_logit_max_kernel(const float* __restrict__ xl, const float* __restrict__ xr,
                                      const int* __restrict__ src, const int* __restrict__ dst,
                                      const float* __restrict__ ea, const float* __restrict__ We,
                                      const float* __restrict__ att,
                                      float* __restrict__ logit, float* __restrict__ m,
                                      int E, int H) {
  const int widx = threadIdx.x >> 5;
  const int lane = threadIdx.x & 31;
  const int t = blockIdx.x * (blockDim.x >> 5) + widx;
  if (t >= E * H) return;
  const int e = t / H;
  const int h = t - e * H;
  const int HC = H * 64;
  const int s = src[e];
  const int d = dst[e];
  const float w = ea[e];
  const v2f* xlp = (const v2f*)(xl + (size_t)s * HC + h * 64);
  const v2f* xrp = (const v2f*)(xr + (size_t)d * HC + h * 64);
  const v2f* wep = (const v2f*)(We + h * 64);
  const v2f* atp = (const v2f*)(att + h * 64);

  const v2f zl = xlp[lane], zr = xrp[lane], we = wep[lane], at = atp[lane];
  float z0 = zl.x + zr.x + w * we.x;
  float z1 = zl.y + zr.y + w * we.y;
  z0 = (z0 > 0.0f) ? z0 : 0.2f * z0;  // leaky_relu(., 0.2)
  z1 = (z1 > 0.0f) ? z1 : 0.2f * z1;
  float acc = z0 * at.x + z1 * at.y;

  for (int off = 16; off > 0; off >>= 1) acc += __shfl_xor(acc, off, 32);
  if (lane == 0) {
    logit[t] = acc;
    atomicMaxFloat(&m[(size_t)d * H + h], acc);
  }
}

__global__ void edge_exp_sum_kernel(const float* __restrict__ logit, const int* __restrict__ dst,
                                    const float* __restrict__ m, float* __restrict__ ex,
                                    float* __restrict__ den, int E, int H) {
  const int i = blockIdx.x * blockDim.x + threadIdx.x;
  if (i >= E * H) return;
  const int e = i / H;
  const int h = i - e * H;
  const int d = dst[e];
  const float v = expf(logit[i] - m[(size_t)d * H + h]);
  ex[i] = v;
  atomicAdd(&den[(size_t)d * H + h], v);
}

__global__ void edge_aggregate_kernel(const float* __restrict__ xl, const int* __restrict__ src,
                                      const int* __restrict__ dst, const float* __restrict__ ex,
                                      const float* __restrict__ den, float* __restrict__ acc,
                                      int E, int H) {
  const int widx = threadIdx.x >> 5;
  const int lane = threadIdx.x & 31;
  const int t = blockIdx.x * (blockDim.x >> 5) + widx;
  if (t >= E * H) return;
  const int e = t / H;
  const int h = t - e * H;
  const int HC = H * 64;
  const int s = src[e];
  const int d = dst[e];
  const float a = ex[t] / (den[(size_t)d * H + h] + 1e-16f);
  const v2f xv = ((const v2f*)(xl + (size_t)s * HC + h * 64))[lane];
  float* ap = acc + (size_t)d * HC + h * 64 + 2 * lane;
  atomicAdd(ap + 0, xv.x * a);
  atomicAdd(ap + 1, xv.y * a);
}

// buf[i] = maybe_relu( scale * (buf[i] + b1[f] + b2[f]) ),  f = i & hc_mask
__global__ void epilogue_kernel(float* __restrict__ buf, const float* __restrict__ b1,
                                const float* __restrict__ b2, float scale, int do_relu,
                                int n, int hc_mask) {
  const int i = blockIdx.x * blockDim.x + threadIdx.x;
  if (i >= n) return;
  const int f = i & hc_mask;
  float v = buf[i] + b1[f] + (b2 ? b2[f] : 0.0f);
  v *= scale;
  if (do_relu) v = fmaxf(v, 0.0f);
  buf[i] = v;
}

// ----------------------------------------------------------- host helpers

struct GatP {
  const float *We, *Wl, *Wr, *att, *bias, *bl, *br;
};

static GatP gat_params(void* const* d_in, int base) {
  GatP p;
  p.We   = (const float*)d_in[base + 0];
  p.Wl   = (const float*)d_in[base + 1];
  p.Wr   = (const float*)d_in[base + 2];
  p.att  = (const float*)d_in[base + 3];
  p.bias = (const float*)d_in[base + 4];
  p.bl   = (const float*)d_in[base + 5];
  p.br   = (const float*)d_in[base + 6];
  return p;
}

static inline int cdiv(int a, int b) { return (a + b - 1) / b; }

template <int K, int H>
static void run_gatv2(const float* xsrc, const float* xdst,
                      const int* src, const int* dst, const float* ea,
                      const GatP& p, int n_src, int n_dst, int E,
                      float* xl, float* xr, float* lg, float* exb, float* mb, float* db,
                      float* acc, hipStream_t stream) {
  constexpr int HC = H * 64;
  // xl = xsrc @ Wl + bl ; xr = xdst @ Wr + br
  gemm_bias_wmma_kernel<K, HC><<<cdiv(n_src, 128), 256, 0, stream>>>(xsrc, p.Wl, p.bl, xl, n_src);
  gemm_bias_wmma_kernel<K, HC><<<cdiv(n_dst, 128), 256, 0, stream>>>(xdst, p.Wr, p.br, xr, n_dst);
  // init segment max (-inf) and segment sum (0)
  const int nd = n_dst * H;
  fill_u32_kernel<<<cdiv(nd, 256), 256, 0, stream>>>((unsigned*)mb, 0xFF800000u, nd);
  fill_u32_kernel<<<cdiv(nd, 256), 256, 0, stream>>>((unsigned*)db, 0u, nd);
  // edge passes
  const int tot = E * H;
  edge_logit_max_kernel<<<cdiv(tot, 8), 256, 0, stream>>>(xl, xr, src, dst, ea, p.We, p.att,
                                                          lg, mb, E, H);
  edge_exp_sum_kernel<<<cdiv(tot, 256), 256, 0, stream>>>(lg, dst, mb, exb, db, E, H);
  edge_aggregate_kernel<<<cdiv(tot, 8), 256, 0, stream>>>(xl, src, dst, exb, db, acc, E, H);
}

extern "C" void kernel_launch(void* const* d_in, const int* in_sizes, int n_in,
                              void* d_out, int out_size, void* d_ws, size_t ws_size,
                              hipStream_t stream) {
  if (n_in < 80) return;

  // ---- raw inputs
  const float* x_user = (const float*)d_in[0];
  const float* x_poi  = (const float*)d_in[1];
  const float* x_cate = (const float*)d_in[2];
  const int *src_u2p = (const int*)d_in[3],  *dst_u2p = (const int*)d_in[4];
  const float* ea_u2p = (const float*)d_in[5];
  const int *src_p2u = (const int*)d_in[6],  *dst_p2u = (const int*)d_in[7];
  const float* ea_p2u = (const float*)d_in[8];
  const int *src_p2c = (const int*)d_in[9],  *dst_p2c = (const int*)d_in[10];
  const float* ea_p2c = (const float*)d_in[11];
  const int *src_c2p = (const int*)d_in[12], *dst_c2p = (const int*)d_in[13];
  const float* ea_c2p = (const float*)d_in[14];

  // ---- params (jax pytree leaf order: dict keys sorted recursively)
  const float* cate_emb = (const float*)d_in[15];
  const float* cate_W   = (const float*)d_in[16];
  const float* cate_b   = (const float*)d_in[17];
  GatP c1_c2p = gat_params(d_in, 18);
  GatP c1_p2c = gat_params(d_in, 25);
  GatP c1_p2u = gat_params(d_in, 32);
  GatP c1_u2p = gat_params(d_in, 39);
  GatP c2_c2p = gat_params(d_in, 46);
  GatP c2_p2c = gat_params(d_in, 53);
  GatP c2_p2u = gat_params(d_in, 60);
  GatP c2_u2p = gat_params(d_in, 67);
  const float* poi_emb  = (const float*)d_in[74];
  const float* poi_W    = (const float*)d_in[75];
  const float* poi_b    = (const float*)d_in[76];
  const float* user_emb = (const float*)d_in[77];
  const float* user_W   = (const float*)d_in[78];
  const float* user_b   = (const float*)d_in[79];

  // ---- workspace layout (floats)
  float* ws  = (float*)d_ws;
  float* n_u = ws;                        // NUSR*64
  float* n_p = n_u + (size_t)NUSR * 64;   // NPOI*64
  float* n_c = n_p + (size_t)NPOI * 64;   // NCAT*64
  float* h_u = n_c + (size_t)NCAT * 64;   // NUSR*128
  float* h_p = h_u + (size_t)NUSR * 128;  // NPOI*128
  float* h_c = h_p + (size_t)NPOI * 128;  // NCAT*128
  float* xl  = h_c + (size_t)NCAT * 128;  // NUSR*128
  float* xr  = xl + (size_t)NUSR * 128;   // NUSR*128
  float* lg  = xr + (size_t)NUSR * 128;   // E_UP*2
  float* exb = lg + (size_t)E_UP * 2;     // E_UP*2
  float* mb  = exb + (size_t)E_UP * 2;    // NUSR*2
  float* db  = mb + (size_t)NUSR * 2;     // NUSR*2

  // ---- layer 0: node feature + embedding
  node_embed_kernel<<<cdiv(NUSR * HID, 256), 256, 0, stream>>>(x_user, user_W, user_b, user_emb,
                                                               n_u, NUSR, 2);
  node_embed_kernel<<<cdiv(NPOI * HID, 256), 256, 0, stream>>>(x_poi, poi_W, poi_b, poi_emb,
                                                               n_p, NPOI, 3);
  node_embed_kernel<<<cdiv(NCAT * HID, 256), 256, 0, stream>>>(x_cate, cate_W, cate_b, cate_emb,
                                                               n_c, NCAT, 1);

  // ---- layer 1: H=2, C=64 (HC=128), fin=64
  fill_u32_kernel<<<cdiv(NUSR * 128, 256), 256, 0, stream>>>((unsigned*)h_u, 0u, NUSR * 128);
  fill_u32_kernel<<<cdiv(NPOI * 128, 256), 256, 0, stream>>>((unsigned*)h_p, 0u, NPOI * 128);
  fill_u32_kernel<<<cdiv(NCAT * 128, 256), 256, 0, stream>>>((unsigned*)h_c, 0u, NCAT * 128);

  run_gatv2<64, 2>(n_p, n_u, src_p2u, dst_p2u, ea_p2u, c1_p2u, NPOI, NUSR, E_UP,
                   xl, xr, lg, exb, mb, db, h_u, stream);
  run_gatv2<64, 2>(n_u, n_p, src_u2p, dst_u2p, ea_u2p, c1_u2p, NUSR, NPOI, E_UP,
                   xl, xr, lg, exb, mb, db, h_p, stream);
  run_gatv2<64, 2>(n_c, n_p, src_c2p, dst_c2p, ea_c2p, c1_c2p, NCAT, NPOI, E_PC,
                   xl, xr, lg, exb, mb, db, h_p, stream);
  run_gatv2<64, 2>(n_p, n_c, src_p2c, dst_p2c, ea_p2c, c1_p2c, NPOI, NCAT, E_PC,
                   xl, xr, lg, exb, mb, db, h_c, stream);

  epilogue_kernel<<<cdiv(NUSR * 128, 256), 256, 0, stream>>>(h_u, c1_p2u.bias, nullptr, 1.0f, 1,
                                                             NUSR * 128, 127);
  epilogue_kernel<<<cdiv(NPOI * 128, 256), 256, 0, stream>>>(h_p, c1_u2p.bias, c1_c2p.bias, 0.5f, 1,
                                                             NPOI * 128, 127);
  epilogue_kernel<<<cdiv(NCAT * 128, 256), 256, 0, stream>>>(h_c, c1_p2c.bias, nullptr, 1.0f, 1,
                                                             NCAT * 128, 127);

  // ---- layer 2: H=1, C=64 (HC=64), fin=128 -> outputs straight into d_out
  float* out_u = (float*)d_out;              // NUSR*64
  float* out_p = out_u + (size_t)NUSR * 64;  // NPOI*64
  float* out_c = out_p + (size_t)NPOI * 64;  // NCAT*64
  fill_u32_kernel<<<cdiv(out_size, 256), 256, 0, stream>>>((unsigned*)d_out, 0u, out_size);

  run_gatv2<128, 1>(h_p, h_u, src_p2u, dst_p2u, ea_p2u, c2_p2u, NPOI, NUSR, E_UP,
                    xl, xr, lg, exb, mb, db, out_u, stream);
  run_gatv2<128, 1>(h_u, h_p, src_u2p, dst_u2p, ea_u2p, c2_u2p, NUSR, NPOI, E_UP,
                    xl, xr, lg, exb, mb, db, out_p, stream);
  run_gatv2<128, 1>(h_c, h_p, src_c2p, dst_c2p, ea_c2p, c2_c2p, NCAT, NPOI, E_PC,
                    xl, xr, lg, exb, mb, db, out_p, stream);
  run_gatv2<128, 1>(h_p, h_c, src_p2c, dst_p2c, ea_p2c, c2_p2c, NPOI, NCAT, E_PC,
                    xl, xr, lg, exb, mb, db, out_c, stream);

  epilogue_kernel<<<cdiv(NUSR * 64, 256), 256, 0, stream>>>(out_u, c2_p2u.bias, nullptr, 1.0f, 0,
                                                            NUSR * 64, 63);
  epilogue_kernel<<<cdiv(NPOI * 64, 256), 256, 0, stream>>>(out_p, c2_u2p.bias, c2_c2p.bias, 0.5f, 0,
                                                            NPOI * 64, 63);
  epilogue_kernel<<<cdiv(NCAT * 64, 256), 256, 0, stream>>>(out_c, c2_p2c.bias, nullptr, 1.0f, 0,
                                                            NCAT * 64, 63);
}